// GemmaWithRegressionHead_13597866459574
// MI455X (gfx1250) — compile-verified
//
#include <hip/hip_runtime.h>
#include <math.h>

#define KL_ALPHA 0.3f

// -------- CDNA5 WMMA wave32 f32 sum-reduction --------
// V_WMMA_F32_16X16X4_F32: A is 16x4 f32 (2 VGPRs), B is 4x16 f32 (2 VGPRs),
// C/D is 16x16 f32 (8 VGPRs). Place lane value v into A VGPR0 only:
//   A[m,0] = v[lane m] (lanes 0-15), A[m,2] = v[lane m+16] (lanes 16-31), A[m,1]=A[m,3]=0.
// With B = all ones: D[m,n] = v[m] + v[m+16] for every n.
// D layout: VGPR j, lanes 0-15 hold M=j; lanes 16-31 hold M=j+8. So
// sum_j d[j] = sum of half the rowsums; one shfl_xor(16) completes the wave sum.
typedef float v2f __attribute__((ext_vector_type(2)));
typedef float v8f __attribute__((ext_vector_type(8)));

__device__ __forceinline__ float wave_sum32(float v) {
    v2f a; a[0] = v;    a[1] = 0.0f;
    v2f b; b[0] = 1.0f; b[1] = 1.0f;
    v8f c = {0.f, 0.f, 0.f, 0.f, 0.f, 0.f, 0.f, 0.f};
    v8f d = __builtin_amdgcn_wmma_f32_16x16x4_f32(false, a, false, b,
                                                  (short)0, c, false, false);
    float s = d[0] + d[1] + d[2] + d[3] + d[4] + d[5] + d[6] + d[7];
    s += __shfl_xor(s, 16, 32);
    return s;  // full 32-lane sum broadcast to all lanes
}

// -------- Kernel 1: per-row logsumexp + fused KL dot, single HBM pass --------
// One block per (b,s) row, s in [0,512). logits rows have stride 32000 with
// 513 rows per batch (row 512 is unused by the reference: lp[:, :-1]).
__global__ __launch_bounds__(256) void row_kernel(
    const float* __restrict__ logits,
    const float* __restrict__ ref_logits,
    const int*   __restrict__ input_ids,
    float4*      __restrict__ rowvals)   // per row: {surprisal, kl_tok, g, gr}
{
    const int row = blockIdx.x;          // 0..2047
    const int b   = row >> 9;
    const int s   = row & 511;
    const int tid = threadIdx.x;
    const size_t off = ((size_t)(b * 513 + s)) * 32000u;
    const float4* __restrict__ x4 = (const float4*)(logits + off);
    const float4* __restrict__ r4 = (const float4*)(ref_logits + off);

    // per-thread online logsumexp state for x and r; W rides along with (mr,Sr)
    float mx = -3.402823466e38f, mr = -3.402823466e38f;
    float Sx = 0.f, Sr = 0.f, W = 0.f;

    for (int i = tid; i < 8000; i += 256) {               // 32000 f32 = 8000 f32x4
        if (i + 256 < 8000) {                             // stream-ahead hint
            __builtin_prefetch((const void*)(x4 + i + 256), 0, 1);
            __builtin_prefetch((const void*)(r4 + i + 256), 0, 1);
        }
        float4 xv = x4[i];
        float4 rv = r4[i];
        #pragma unroll
        for (int cc = 0; cc < 4; ++cc) {
            float x = (&xv.x)[cc];
            float r = (&rv.x)[cc];
            if (x <= mx) {                                // common path: 1 exp
                Sx += __expf(x - mx);
            } else {                                      // rare: new max
                Sx = Sx * __expf(mx - x) + 1.0f;
                mx = x;
            }
            float dxr = r - x;
            if (r <= mr) {
                float e = __expf(r - mr);
                Sr += e;
                W  += e * dxr;
            } else {
                float e = __expf(mr - r);
                Sr = Sr * e + 1.0f;
                W  = W  * e + dxr;
                mr = r;
            }
        }
    }

    __shared__ float smax[2][8];
    __shared__ float ssum[3][8];
    const int lane = tid & 31, wave = tid >> 5;

    // block-wide max (shuffle tree within wave, LDS across 8 waves)
    float wmx = mx, wmr = mr;
    #pragma unroll
    for (int o = 16; o; o >>= 1) {
        wmx = fmaxf(wmx, __shfl_xor(wmx, o, 32));
        wmr = fmaxf(wmr, __shfl_xor(wmr, o, 32));
    }
    if (lane == 0) { smax[0][wave] = wmx; smax[1][wave] = wmr; }
    __syncthreads();
    float MX = smax[0][0], MR = smax[1][0];
    #pragma unroll
    for (int w = 1; w < 8; ++w) {
        MX = fmaxf(MX, smax[0][w]);
        MR = fmaxf(MR, smax[1][w]);
    }

    // rescale per-thread partials to the block max, then plain-sum via WMMA
    Sx *= __expf(mx - MX);
    float sc = __expf(mr - MR);
    Sr *= sc;
    W  *= sc;

    float sxw = wave_sum32(Sx);
    float srw = wave_sum32(Sr);
    float ww  = wave_sum32(W);
    if (lane == 0) { ssum[0][wave] = sxw; ssum[1][wave] = srw; ssum[2][wave] = ww; }
    __syncthreads();

    if (tid == 0) {
        float TSx = 0.f, TSr = 0.f, TW = 0.f;
        #pragma unroll
        for (int w = 0; w < 8; ++w) {
            TSx += ssum[0][w]; TSr += ssum[1][w]; TW += ssum[2][w];
        }
        float lse_x = MX + __logf(TSx);
        float lse_r = MR + __logf(TSr);

        const float* __restrict__ xr = logits + off;
        const float* __restrict__ rr = ref_logits + off;
        int   id_cur = input_ids[b * 512 + s];
        float x_cur  = xr[id_cur];
        float g = 0.f, gr = 0.f;
        if (s < 511) {                                    // g/gr exist for s<S-1
            int id_next = input_ids[b * 512 + s + 1];
            g  = xr[id_next] - lse_x;
            gr = rr[id_next] - lse_r;
        }
        float4 o;
        o.x = lse_x - x_cur;                              // surprisal[b,s]
        o.y = TW / TSr + (lse_x - lse_r);                 // kl_tok[b,s]
        o.z = g;
        o.w = gr;
        rowvals[row] = o;
    }
}

// -------- Kernel 2: scatter-add by word_ids, regression head, per-batch partials ----
__global__ __launch_bounds__(512) void head_kernel(
    const float4* __restrict__ rowvals,
    const int*    __restrict__ word_ids,
    const int*    __restrict__ word_lengths,
    const float*  __restrict__ luf,
    const float*  __restrict__ labels,
    const int*    __restrict__ attention_mask,
    const float*  __restrict__ W_head,
    const float*  __restrict__ b_head,
    float*        __restrict__ preds_out,   // d_out + 1, length 2048
    float*        __restrict__ partials)    // 4 batches * 8 floats
{
    const int b = blockIdx.x;
    const int t = threadIdx.x;
    const int base = b * 512;

    __shared__ float summed[512];
    summed[t] = 0.f;
    __syncthreads();

    float4 rv = rowvals[base + t];
    int  wid  = word_ids[base + t];
    bool msk  = (wid != -1);
    if (msk && wid >= 0 && wid < 512) atomicAdd(&summed[wid], rv.x);
    __syncthreads();

    // regression head
    float ss    = summed[t];
    bool  valid = (wid >= 0) && (wid < 512);
    int   safe  = wid < 0 ? 0 : (wid > 511 ? 511 : wid);
    int   len   = word_lengths[base + safe];
    valid = valid && (len != -1);
    float exp_len  = valid ? (float)len : 0.f;
    float exp_freq = valid ? luf[base + safe] : 0.f;
    float pred = ss * W_head[0] + exp_len * W_head[1] + exp_freq * W_head[2] + b_head[0];
    preds_out[base + t] = pred;

    // partial-sum quantities
    float label = labels[base + t];
    float mw    = (label != -1.0f) ? 1.f : 0.f;
    float dd    = label - pred;
    float klm   = (t == 0) ? 0.f : ((word_ids[base + t - 1] != -1) ? 1.f : 0.f);
    float am    = (t < 511) ? (float)attention_mask[base + t + 1] : 0.f;
    float q[7];
    q[0] = dd * dd * mw;   // mse numerator
    q[1] = mw;             // mse denominator
    q[2] = rv.y * klm;     // kl numerator
    q[3] = klm;            // kl denominator
    q[4] = rv.z * am;      // sum g * ans_mask
    q[5] = rv.w * am;      // sum gr * ans_mask
    q[6] = am;             // sum ans_mask

    __shared__ float red[7][16];
    const int lane = t & 31, wave = t >> 5;
    #pragma unroll
    for (int k = 0; k < 7; ++k) {
        float sv = wave_sum32(q[k]);
        if (lane == 0) red[k][wave] = sv;
    }
    __syncthreads();
    if (t == 0) {
        #pragma unroll
        for (int k = 0; k < 7; ++k) {
            float acc = 0.f;
            #pragma unroll
            for (int w = 0; w < 16; ++w) acc += red[k][w];
            partials[b * 8 + k] = acc;
        }
    }
}

// -------- Kernel 3: finalize scalars --------
__global__ void final_kernel(const float* __restrict__ partials,
                             float* __restrict__ out)
{
    if (threadIdx.x != 0 || blockIdx.x != 0) return;
    float a[7] = {0.f, 0.f, 0.f, 0.f, 0.f, 0.f, 0.f};
    for (int b = 0; b < 4; ++b)
        for (int k = 0; k < 7; ++k) a[k] += partials[b * 8 + k];
    float mse   = a[0] / a[1];
    float kl    = a[2] / a[3];
    float avg   = a[4] / a[6];
    float delta = avg - a[5] / a[6];
    out[0]    = mse + KL_ALPHA * kl;   // loss (MSE_WEIGHT = 1)
    out[2049] = mse;
    out[2050] = kl;
    out[2051] = avg;
    out[2052] = delta;
}

extern "C" void kernel_launch(void* const* d_in, const int* in_sizes, int n_in,
                              void* d_out, int out_size, void* d_ws, size_t ws_size,
                              hipStream_t stream) {
    const float* logits         = (const float*)d_in[0];
    const float* ref_logits     = (const float*)d_in[1];
    const int*   input_ids      = (const int*)  d_in[2];
    const int*   attention_mask = (const int*)  d_in[3];
    const int*   word_ids       = (const int*)  d_in[4];
    const int*   word_lengths   = (const int*)  d_in[5];
    const float* luf            = (const float*)d_in[6];
    const float* labels         = (const float*)d_in[7];
    const float* W_head         = (const float*)d_in[8];
    const float* b_head         = (const float*)d_in[9];

    float*  out      = (float*)d_out;
    float4* rowvals  = (float4*)d_ws;                                  // 2048 * 16 B
    float*  partials = (float*)((char*)d_ws + 2048 * sizeof(float4));  // 32 floats

    row_kernel<<<2048, 256, 0, stream>>>(logits, ref_logits, input_ids, rowvals);
    head_kernel<<<4, 512, 0, stream>>>(rowvals, word_ids, word_lengths, luf, labels,
                                       attention_mask, W_head, b_head,
                                       out + 1, partials);
    final_kernel<<<1, 32, 0, stream>>>(partials, out);
}